// Son_swapnet_16544214024204
// MI455X (gfx1250) — compile-verified
//
#include <hip/hip_runtime.h>

typedef __attribute__((ext_vector_type(2))) float v2f;
typedef __attribute__((ext_vector_type(8))) float v8f;

#define C_NODES 256
#define TILE 16
#define WAVES_PER_WG 8

// out[b,i] = sum_p g(o1[b,p]-o1[b,i]) * W[p,i]
// g(d) = sign(d)*leaky_relu(c0*|d| + c1*|d|^2 + c2*|d|^3 + c3*|d|^4)
// W[p,q] = w3[e(min,max)] (symmetric), W[i,i] irrelevant since g(0)=0.
__global__ __launch_bounds__(256) void son_swapnet_kernel(
    const float* __restrict__ x,
    const float* __restrict__ w1,
    const float* __restrict__ w2,
    const float* __restrict__ w3,
    float* __restrict__ out)
{
    __shared__ float sW[C_NODES * TILE];        // W[row, cb+n] slice: 16 KB
    __shared__ float sO[WAVES_PER_WG][C_NODES]; // o1 row per wave:   8 KB

    const int tid  = threadIdx.x;
    const int lane = tid & 31;
    const int w    = tid >> 5;
    const int cbg  = blockIdx.x * TILE;             // column-tile base (node index)
    const int b    = blockIdx.y * WAVES_PER_WG + w; // batch for this wave

    // ---- expand w3 -> symmetric W slice [256 x 16] in LDS ----
    #pragma unroll
    for (int it = 0; it < 16; ++it) {
        int flat = it * 256 + tid;   // 0..4095
        int row  = flat >> 4;
        int n    = flat & 15;
        int col  = cbg + n;
        float wv = 0.0f;
        if (row != col) {
            int p = row < col ? row : col;
            int q = row < col ? col : row;
            int e = p * (C_NODES - 1) - ((p * (p - 1)) >> 1) + (q - p - 1);
            wv = w3[e];
        }
        sW[flat] = wv;
    }

    // ---- load o1 = x * w1 row for this wave's batch ----
    #pragma unroll
    for (int k = 0; k < C_NODES / 32; ++k) {
        int c = lane + k * 32;
        sO[w][c] = x[b * C_NODES + c] * w1[c];
    }
    __syncthreads();

    const float c0 = w2[0], c1 = w2[1], c2 = w2[2], c3 = w2[3];

    const int  hi = lane >> 4;
    const int  nn = lane & 15;
    const bool lo = (lane < 16);

    // Loop-invariant WMMA operands.
    // A (16x4): col K=0 = o1[rb+M], K=1 = 1, K=2..3 = 0 (zeros live in lanes 16-31)
    // B (4x16): row K=0 = 1,        K=1 = -o1[cb+N], K=2..3 = 0
    // => D[M,N] = o1[rb+M] - o1[cb+N]
    v2f bmat;
    {
        float bl = sO[w][cbg + nn];                 // unconditional LDS read (in range)
        bmat.x = lo ? 1.0f : 0.0f;
        bmat.y = lo ? -bl : 0.0f;
    }
    const float a_k1 = lo ? 1.0f : 0.0f;

    // Preload all 16 A-column values into registers: kills the per-tile
    // guarded ds_load + dscnt wait in front of each wmma.
    float a0[16];
    #pragma unroll
    for (int t = 0; t < 16; ++t) {
        float vl = sO[w][t * TILE + nn];            // unconditional LDS read
        a0[t] = lo ? vl : 0.0f;
    }

    float acc = 0.0f;  // per-lane: sum over M-half and all row tiles of T[M, N=nn]
    #pragma unroll
    for (int t = 0; t < 16; ++t) {
        const int rb = t * TILE;
        v2f amat;
        amat.x = a0[t];
        amat.y = a_k1;
        v8f cz = {};
        // v_wmma_f32_16x16x4_f32: rank-2 outer-difference tile in one op
        v8f d = __builtin_amdgcn_wmma_f32_16x16x4_f32(
            false, amat, false, bmat, (short)0, cz, false, false);
        #pragma unroll
        for (int v = 0; v < 8; ++v) {
            float dv   = d[v];                       // element (M = v + 8*hi, N = nn)
            float ad   = fabsf(dv);
            float poly = ad * fmaf(ad, fmaf(ad, fmaf(ad, c3, c2), c1), c0);
            // leaky_relu(v) == max(v, 0.01*v) for all v
            float lr   = fmaxf(poly, 0.01f * poly);
            // sign(dv)*lr via sign-bit xor (avoids VOPC; g(+-0)=0 since poly(0)=0)
            float g    = __uint_as_float(__float_as_uint(lr) ^
                                         (__float_as_uint(dv) & 0x80000000u));
            int   row  = rb + v + 8 * hi;
            acc = fmaf(g, sW[row * TILE + nn], acc);
        }
    }
    // combine the two M-halves (lanes L and L+16 hold complementary partial sums)
    acc += __shfl_xor(acc, 16, 32);
    if (lo) out[b * C_NODES + cbg + nn] = acc;
}

extern "C" void kernel_launch(void* const* d_in, const int* in_sizes, int n_in,
                              void* d_out, int out_size, void* d_ws, size_t ws_size,
                              hipStream_t stream) {
    (void)in_sizes; (void)n_in; (void)d_ws; (void)ws_size; (void)out_size;
    const float* x  = (const float*)d_in[0];  // [1024, 256]
    const float* w1 = (const float*)d_in[1];  // [256]
    const float* w2 = (const float*)d_in[2];  // [4]
    const float* w3 = (const float*)d_in[3];  // [32640]
    // d_in[4] (diff_indices, 33 MB) intentionally unused: aggregation done analytically.
    float* out = (float*)d_out;               // [1024, 256, 1]

    dim3 block(256);
    dim3 grid(C_NODES / TILE, 1024 / WAVES_PER_WG);  // 16 x 128
    son_swapnet_kernel<<<grid, block, 0, stream>>>(x, w1, w2, w3, out);
}